// Voxelizer_31731218383464
// MI455X (gfx1250) — compile-verified
//
#include <hip/hip_runtime.h>
#include <hip/hip_bf16.h>
#include <stdint.h>

// ---------------- problem constants (from reference) ----------------
#define NGRID      16
#define NVOX       4096            // 16^3
#define NCOMP      10              // cnt, sx,sy,sz, xx,xy,xz,yy,yz,zz
#define MIN_PTS    3.0f
#define VEPS       1e-6f
#define NB         8               // batch (setup_inputs)

// ---------------- tiling / launch config ----------------
#define TILE       1024            // points per staged LDS tile
#define BLK2       768             // accumulate threads: TILE*3/4 == 768 chunks (1 async op/thread)
#define BPC        32              // blocks per cloud, accumulate pass
#define BLK1       256
#define B1PC       16              // blocks per cloud, minmax pass

// ws layout (32-bit units): [0..48) encoded bounds, [64 ..) stats accumulators
#define WS_STATS_OFF 64

#define AS1 __attribute__((address_space(1)))
#define AS3 __attribute__((address_space(3)))

// Pointee type of the async-to-LDS b128 builtin (from hipcc diagnostic):
// "__attribute__((__vector_size__(4 * sizeof(int)))) int __device__ *"
typedef int v4i __attribute__((vector_size(16)));

// gfx1250 async-to-LDS path (ASYNCcnt), guarded so we compile on any toolchain
#if defined(__has_builtin)
#  if __has_builtin(__builtin_amdgcn_global_load_async_to_lds_b128) && \
      __has_builtin(__builtin_amdgcn_s_wait_asynccnt)
#    define USE_ASYNC 1
#  endif
#endif
#ifndef USE_ASYNC
#  define USE_ASYNC 0
#endif

// Relaxed, workgroup-scope FP add on LDS -> native ds_add_f32 (no CAS loop)
__device__ __forceinline__ void lds_fadd(float* p, float v) {
  __hip_atomic_fetch_add(p, v, __ATOMIC_RELAXED, __HIP_MEMORY_SCOPE_WORKGROUP);
}

// order-preserving float <-> uint encoding for atomic min/max
__device__ __forceinline__ unsigned fenc(float f) {
  unsigned u = __float_as_uint(f);
  return (u & 0x80000000u) ? ~u : (u | 0x80000000u);
}
__device__ __forceinline__ float fdec(unsigned u) {
  return __uint_as_float((u & 0x80000000u) ? (u & 0x7fffffffu) : ~u);
}

// ---------------- kernel 0: init workspace ----------------
__global__ void vox_init(unsigned* __restrict__ ws_u32, float* __restrict__ stats,
                         int statTotal) {
  int i = blockIdx.x * blockDim.x + threadIdx.x;
  int stride = gridDim.x * blockDim.x;
  for (int k = i; k < NB * 6; k += stride)
    ws_u32[k] = ((k % 6) < 3) ? 0xFFFFFFFFu : 0u;   // min slots / max slots
  for (int k = i; k < statTotal; k += stride)
    stats[k] = 0.0f;
}

// ---------------- kernel 1: per-cloud min/max ----------------
__global__ void __launch_bounds__(BLK1) vox_minmax(const float* __restrict__ pts,
                                                   unsigned* __restrict__ bounds,
                                                   int N) {
  const int b   = blockIdx.x / B1PC;
  const int sub = blockIdx.x % B1PC;
  const float* base = pts + (size_t)b * N * 3;
  const int tid    = sub * BLK1 + threadIdx.x;
  const int stride = B1PC * BLK1;

  float mnx = 3.4e38f, mny = 3.4e38f, mnz = 3.4e38f;
  float mxx = -3.4e38f, mxy = -3.4e38f, mxz = -3.4e38f;
  for (int p = tid; p < N; p += stride) {
    __builtin_prefetch(base + 3 * (size_t)(p + 4 * stride), 0, 1);  // global_prefetch_b8
    float3 v = *(const float3*)(base + 3 * (size_t)p);              // global_load_b96
    mnx = fminf(mnx, v.x); mxx = fmaxf(mxx, v.x);
    mny = fminf(mny, v.y); mxy = fmaxf(mxy, v.y);
    mnz = fminf(mnz, v.z); mxz = fmaxf(mxz, v.z);
  }
  // wave32 reduction
  for (int off = 16; off > 0; off >>= 1) {
    mnx = fminf(mnx, __shfl_xor(mnx, off));
    mny = fminf(mny, __shfl_xor(mny, off));
    mnz = fminf(mnz, __shfl_xor(mnz, off));
    mxx = fmaxf(mxx, __shfl_xor(mxx, off));
    mxy = fmaxf(mxy, __shfl_xor(mxy, off));
    mxz = fmaxf(mxz, __shfl_xor(mxz, off));
  }
  if ((threadIdx.x & 31) == 0) {
    unsigned* bb = bounds + b * 6;
    atomicMin(bb + 0, fenc(mnx));
    atomicMin(bb + 1, fenc(mny));
    atomicMin(bb + 2, fenc(mnz));
    atomicMax(bb + 3, fenc(mxx));
    atomicMax(bb + 4, fenc(mxy));
    atomicMax(bb + 5, fenc(mxz));
  }
}

// ---------------- kernel 2: voxel accumulation ----------------
__device__ __forceinline__ void issueTile(const float* gsrc, float* ldst, int tid) {
#if USE_ASYNC
  // TILE*12 bytes, 16 B per lane, chunks == BLK2 -> exactly one async op per thread
  AS1 v4i* g = (AS1 v4i*)gsrc;
  AS3 v4i* l = (AS3 v4i*)ldst;
  __builtin_amdgcn_global_load_async_to_lds_b128(g + tid, l + tid, 0, 0);
#else
  ((float4*)ldst)[tid] = ((const float4*)gsrc)[tid];
#endif
}

__device__ __forceinline__ void accumPoint(float* sstat, float x, float y, float z,
                                           float lox, float loy, float loz,
                                           float ex, float ey, float ez) {
  int ix = (int)floorf((x - lox) / ex * (float)NGRID);
  int iy = (int)floorf((y - loy) / ey * (float)NGRID);
  int iz = (int)floorf((z - loz) / ez * (float)NGRID);
  ix = min(max(ix, 0), NGRID - 1);
  iy = min(max(iy, 0), NGRID - 1);
  iz = min(max(iz, 0), NGRID - 1);
  int vid = (ix * NGRID + iy) * NGRID + iz;
  lds_fadd(&sstat[vid],            1.0f);   // ds_add_f32
  lds_fadd(&sstat[NVOX     + vid], x);
  lds_fadd(&sstat[2 * NVOX + vid], y);
  lds_fadd(&sstat[3 * NVOX + vid], z);
  lds_fadd(&sstat[4 * NVOX + vid], x * x);
  lds_fadd(&sstat[5 * NVOX + vid], x * y);
  lds_fadd(&sstat[6 * NVOX + vid], x * z);
  lds_fadd(&sstat[7 * NVOX + vid], y * y);
  lds_fadd(&sstat[8 * NVOX + vid], y * z);
  lds_fadd(&sstat[9 * NVOX + vid], z * z);
}

__global__ void __launch_bounds__(BLK2) vox_accum(const float* __restrict__ pts,
                                                  const unsigned* __restrict__ bounds,
                                                  float* __restrict__ stats, int N) {
  const int b   = blockIdx.x / BPC;
  const int bid = blockIdx.x % BPC;
  const int tid = threadIdx.x;

  extern __shared__ float smem[];
  float* sstat = smem;                       // NCOMP * NVOX floats (160 KB)
  float* tbuf0 = smem + NCOMP * NVOX;        // TILE*3 floats
  float* tbuf1 = tbuf0 + TILE * 3;

  for (int i = tid; i < NCOMP * NVOX; i += BLK2) sstat[i] = 0.0f;

  const unsigned* bb = bounds + b * 6;
  const float lox = fdec(bb[0]), loy = fdec(bb[1]), loz = fdec(bb[2]);
  const float ex = fmaxf(fdec(bb[3]) - lox, VEPS);
  const float ey = fmaxf(fdec(bb[4]) - loy, VEPS);
  const float ez = fmaxf(fdec(bb[5]) - loz, VEPS);

  const float* cpts = pts + (size_t)b * N * 3;
  const int fullTiles = N / TILE;

  int t = bid, cur = 0;
  if (t < fullTiles) issueTile(cpts + (size_t)t * (TILE * 3), tbuf0, tid);
  for (; t < fullTiles; t += BPC) {
    const int tn = t + BPC;
    if (tn < fullTiles) {
      issueTile(cpts + (size_t)tn * (TILE * 3), cur ? tbuf0 : tbuf1, tid);
#if USE_ASYNC
      __builtin_amdgcn_s_wait_asynccnt(1);   // oldest (tile t) landed, next still in flight
#endif
    } else {
#if USE_ASYNC
      __builtin_amdgcn_s_wait_asynccnt(0);
#endif
    }
    __syncthreads();
    const float* tb = cur ? tbuf1 : tbuf0;
    for (int i = tid; i < TILE; i += BLK2) {
      accumPoint(sstat, tb[3 * i], tb[3 * i + 1], tb[3 * i + 2],
                 lox, loy, loz, ex, ey, ez);
    }
    __syncthreads();
    cur ^= 1;
  }
  // tail points (N % TILE), handled by block 0 of each cloud, direct loads
  if (bid == 0) {
    for (int p = fullTiles * TILE + tid; p < N; p += BLK2) {
      float3 v = *(const float3*)(cpts + 3 * (size_t)p);
      accumPoint(sstat, v.x, v.y, v.z, lox, loy, loz, ex, ey, ez);
    }
  }
  __syncthreads();
  // flush partial sums: coalesced, conflict-free global float atomics into L2
  float* gst = stats + (size_t)b * NCOMP * NVOX;
  for (int i = tid; i < NCOMP * NVOX; i += BLK2) {
    float v = sstat[i];
    if (v != 0.0f) unsafeAtomicAdd(&gst[i], v);  // global_atomic_add_f32
  }
}

// ---------------- kernel 3: finalize + exact stable top-K ----------------
__global__ void __launch_bounds__(1024) vox_finalize(const float* __restrict__ stats,
                                                     float* __restrict__ means,
                                                     float* __restrict__ covs, int K) {
  const int b = blockIdx.x;
  __shared__ float score[NVOX];
  const float* st = stats + (size_t)b * NCOMP * NVOX;

  for (int v = threadIdx.x; v < NVOX; v += blockDim.x) {
    float c = st[v];
    score[v] = (c >= MIN_PTS) ? c : -1.0f;
  }
  __syncthreads();

  for (int v = threadIdx.x; v < NVOX; v += blockDim.x) {
    const float si = score[v];
    int rank = 0;
    for (int j = 0; j < NVOX; ++j) {           // stable top_k rank (bijection 0..V-1)
      float sj = score[j];
      rank += (sj > si) || ((sj == si) && (j < v));
    }
    if (rank < K) {
      float cnt = st[v];
      float* m  = means + ((size_t)b * K + rank) * 3;
      float* c9 = covs  + ((size_t)b * K + rank) * 9;
      if (cnt >= MIN_PTS) {
        float denom = fmaxf(cnt, 1.0f);
        float mx = st[NVOX + v] / denom;
        float my = st[2 * NVOX + v] / denom;
        float mz = st[3 * NVOX + v] / denom;
        m[0] = mx; m[1] = my; m[2] = mz;
        float cxy = st[5 * NVOX + v] / denom - mx * my;
        float cxz = st[6 * NVOX + v] / denom - mx * mz;
        float cyz = st[8 * NVOX + v] / denom - my * mz;
        c9[0] = st[4 * NVOX + v] / denom - mx * mx;
        c9[1] = cxy; c9[2] = cxz;
        c9[3] = cxy;
        c9[4] = st[7 * NVOX + v] / denom - my * my;
        c9[5] = cyz;
        c9[6] = cxz; c9[7] = cyz;
        c9[8] = st[9 * NVOX + v] / denom - mz * mz;
      } else {
        m[0] = m[1] = m[2] = 0.0f;
        #pragma unroll
        for (int q = 0; q < 9; ++q) c9[q] = 0.0f;
      }
    }
  }
}

// ---------------- launch ----------------
extern "C" void kernel_launch(void* const* d_in, const int* in_sizes, int n_in,
                              void* d_out, int out_size, void* d_ws, size_t ws_size,
                              hipStream_t stream) {
  (void)n_in; (void)ws_size;
  const float* pts = (const float*)d_in[0];
  const int B = NB;                          // fixed by reference setup_inputs
  const int N = in_sizes[0] / (B * 3);       // 500000
  const int K = out_size / (B * 12);         // 2048 (3 mean + 9 cov floats per dist)

  unsigned* bounds = (unsigned*)d_ws;
  float*    stats  = (float*)d_ws + WS_STATS_OFF;   // needs (64 + B*NCOMP*NVOX)*4 ~= 1.3 MB
  const int statTotal = B * NCOMP * NVOX;

  vox_init<<<256, 256, 0, stream>>>(bounds, stats, statTotal);
  vox_minmax<<<B * B1PC, BLK1, 0, stream>>>(pts, bounds, N);

  const size_t shmem = (size_t)(NCOMP * NVOX + 2 * TILE * 3) * sizeof(float); // 184 KB
  (void)hipFuncSetAttribute((const void*)vox_accum,
                            hipFuncAttributeMaxDynamicSharedMemorySize, (int)shmem);
  vox_accum<<<B * BPC, BLK2, shmem, stream>>>(pts, bounds, stats, N);

  float* means = (float*)d_out;
  float* covs  = means + (size_t)B * K * 3;
  vox_finalize<<<B, 1024, 0, stream>>>(stats, means, covs, K);
}